// NNLoss_48756468744869
// MI455X (gfx1250) — compile-verified
//
#include <hip/hip_runtime.h>
#include <hip/hip_bf16.h>
#include <stdint.h>

// Problem constants (from reference): B=16, C=3, H=256, W=256, 5x5 window, pad=2
#define NN_B 16
#define NN_C 3
#define NN_H 256
#define NN_W 256
#define BIG_F 1e30f

// Tiling: each block computes a 16(h) x 64(w) tile of one image.
#define TH 16
#define TW 64
// Haloed gt tile in LDS: 3 channels x (TH+4)=20 rows x (TW+4)=68 cols
#define LROWS 20
#define LCOLS 68
#define LCH   (LROWS * LCOLS)        // 1360
#define LTOT  (NN_C * LCH)           // 4080 real entries
#define LPAD  4096                   // padded so 256 threads x 16 iters covers it with full EXEC

__global__ __launch_bounds__(256) void nnloss_tile_kernel(
    const float* __restrict__ pred,
    const float* __restrict__ gt,
    float* __restrict__ partials)
{
    __shared__ float tile[LPAD];   // 16 KB haloed gt tile (+16 pad floats)
    __shared__ float red[8];

    const int tid = threadIdx.x;
    const int blk = blockIdx.x;                // 0..1023
    const int b   = blk >> 6;                  // image index
    const int rem = blk & 63;
    const int th  = rem >> 2;                  // tile row index 0..15
    const int tw  = rem & 3;                   // tile col index 0..3
    const int tile_h0 = th * TH;
    const int tile_w0 = tw * TW;

    const float* gtb   = gt   + (size_t)b * NN_C * NN_H * NN_W;
    const float* predb = pred + (size_t)b * NN_C * NN_H * NN_W;
    const uint64_t gt_base = (uint64_t)(uintptr_t)gtb;
    const unsigned lds_base = (unsigned)(uintptr_t)(&tile[0]);

    // ---- Stage haloed gt tile into LDS with CDNA5 async global->LDS copies ----
    // Clamp OOB addresses to a valid element (patched to BIG afterwards) so every
    // lane issues a load with full EXEC; 16 iterations cover the padded 4096 slots.
#pragma unroll
    for (int it = 0; it < 16; ++it) {
        int idx = tid + it * 256;
        int c, ly, lx;
        if (idx < LTOT) {
            c = idx / LCH;
            int r2 = idx - c * LCH;
            ly = r2 / LCOLS;
            lx = r2 - ly * LCOLS;
        } else {            // padding slots: harmless duplicate load, never read
            c = 0; ly = 0; lx = 0;
        }
        int gy = tile_h0 - 2 + ly;
        int gx = tile_w0 - 2 + lx;
        int gyc = min(max(gy, 0), NN_H - 1);
        int gxc = min(max(gx, 0), NN_W - 1);
        unsigned goff = (unsigned)(((c * NN_H + gyc) * NN_W + gxc) * 4);
        unsigned loff = lds_base + (unsigned)idx * 4u;
        asm volatile("global_load_async_to_lds_b32 %0, %1, %2"
                     :: "v"(loff), "v"(goff), "s"(gt_base) : "memory");
    }

    // ---- Issue pred loads BEFORE waiting on the async copy so their global
    // latency overlaps the tile fill ----
    const int r  = tid >> 4;          // output row within tile, 0..15
    const int w0 = (tid & 15) << 2;   // output col group base, 0..60 (16B aligned)

    float p[NN_C][4];
#pragma unroll
    for (int c = 0; c < NN_C; ++c) {
        const float4 pv = *(const float4*)(predb + (size_t)(c * NN_H + (tile_h0 + r)) * NN_W
                                           + (tile_w0 + w0));
        p[c][0] = pv.x; p[c][1] = pv.y; p[c][2] = pv.z; p[c][3] = pv.w;
    }

    asm volatile("s_wait_asynccnt 0" ::: "memory");
    __syncthreads();

    // Patch out-of-image halo entries with the BIG pad value (reference
    // semantics). Only 36 of 64 tile positions touch the border; the branch is
    // scalar-uniform per block, so interior blocks skip the pass entirely.
    const bool border = (tile_h0 == 0) || (tile_h0 == NN_H - TH) ||
                        (tile_w0 == 0) || (tile_w0 == NN_W - TW);
    if (border) {
#pragma unroll
        for (int it = 0; it < 16; ++it) {
            int idx = tid + it * 256;
            if (idx < LTOT) {
                int c = idx / LCH;
                int r2 = idx - c * LCH;
                int ly = r2 / LCOLS;
                int lx = r2 - ly * LCOLS;
                int gy = tile_h0 - 2 + ly;
                int gx = tile_w0 - 2 + lx;
                if (gy < 0 || gy >= NN_H || gx < 0 || gx >= NN_W)
                    tile[idx] = BIG_F;
            }
        }
    }
    __syncthreads();

    // ---- Compute: each thread owns 4 consecutive output pixels ----
    float acc[4];
#pragma unroll
    for (int k = 0; k < 4; ++k) acc[k] = 3.0e38f;

#pragma unroll
    for (int dy = 0; dy < 5; ++dy) {
        // 8 contiguous gt values per channel serve all (4 pixels x 5 dx) combos.
        float g[NN_C][8];
#pragma unroll
        for (int c = 0; c < NN_C; ++c) {
            const int base = c * LCH + (r + dy) * LCOLS + w0;  // 16B-aligned
#pragma unroll
            for (int j = 0; j < 8; ++j) g[c][j] = tile[base + j];
        }
#pragma unroll
        for (int dx = 0; dx < 5; ++dx) {
#pragma unroll
            for (int k = 0; k < 4; ++k) {
                float s = fabsf(g[0][k + dx] - p[0][k])
                        + fabsf(g[1][k + dx] - p[1][k])
                        + fabsf(g[2][k + dx] - p[2][k]);
                acc[k] = fminf(acc[k], s);
            }
        }
    }

    float v = (acc[0] + acc[1]) + (acc[2] + acc[3]);

    // ---- Deterministic block reduction (wave32 shfl tree, then LDS) ----
#pragma unroll
    for (int off = 16; off > 0; off >>= 1)
        v += __shfl_xor(v, off, 32);
    if ((tid & 31) == 0) red[tid >> 5] = v;
    __syncthreads();
    if (tid == 0) {
        float s = 0.0f;
#pragma unroll
        for (int i = 0; i < 8; ++i) s += red[i];
        partials[blk] = s;
    }
}

__global__ __launch_bounds__(256) void nnloss_reduce_kernel(
    const float* __restrict__ partials, float* __restrict__ out)
{
    __shared__ float red[8];
    const int tid = threadIdx.x;
    float v = 0.0f;
#pragma unroll
    for (int i = 0; i < 4; ++i) v += partials[tid + i * 256];
#pragma unroll
    for (int off = 16; off > 0; off >>= 1)
        v += __shfl_xor(v, off, 32);
    if ((tid & 31) == 0) red[tid >> 5] = v;
    __syncthreads();
    if (tid == 0) {
        float s = 0.0f;
#pragma unroll
        for (int i = 0; i < 8; ++i) s += red[i];
        out[0] = s;
    }
}

extern "C" void kernel_launch(void* const* d_in, const int* in_sizes, int n_in,
                              void* d_out, int out_size, void* d_ws, size_t ws_size,
                              hipStream_t stream) {
    const float* pred = (const float*)d_in[0];   // pred_target [16,3,256,256] f32
    const float* gt   = (const float*)d_in[1];   // gt_target   [16,3,256,256] f32
    float* out = (float*)d_out;                  // scalar f32
    float* partials = (float*)d_ws;              // 1024 floats of scratch

    const int nblocks = NN_B * (NN_H / TH) * (NN_W / TW);  // 1024
    nnloss_tile_kernel<<<nblocks, 256, 0, stream>>>(pred, gt, partials);
    nnloss_reduce_kernel<<<1, 256, 0, stream>>>(partials, out);
}